// GVAEEncoder_87505663689257
// MI455X (gfx1250) — compile-verified
//
#include <hip/hip_runtime.h>
#include <hip/hip_bf16.h>

typedef __attribute__((ext_vector_type(2))) float v2f;
typedef __attribute__((ext_vector_type(8))) float v8f;

// ---------------------------------------------------------------------------
// Fill a float buffer with a constant (grid-stride).
// ---------------------------------------------------------------------------
__global__ void k_fill(float* __restrict__ p, size_t n, float v) {
    size_t i = (size_t)blockIdx.x * blockDim.x + threadIdx.x;
    size_t stride = (size_t)gridDim.x * blockDim.x;
    for (; i < n; i += stride) p[i] = v;
}

// ---------------------------------------------------------------------------
// deg[dst[e]] += 1  (self-loop "+1" is pre-filled by k_fill(deg, 1.0f))
// ---------------------------------------------------------------------------
__global__ void k_deg_edges(const int* __restrict__ dst, float* __restrict__ deg, int E) {
    int e = blockIdx.x * blockDim.x + threadIdx.x;
    if (e < E) unsafeAtomicAdd(&deg[dst[e]], 1.0f);
}

// dis = deg > 0 ? rsqrt(deg) : 0
__global__ void k_dis(const float* __restrict__ deg, float* __restrict__ dis, int N) {
    int i = blockIdx.x * blockDim.x + threadIdx.x;
    if (i < N) {
        float d = deg[i];
        dis[i] = (d > 0.0f) ? rsqrtf(d) : 0.0f;
    }
}

// Pack Wcat[k][c] = c<64 ? Wmu[k][c] : Wlv[k][c-64]   (128x128)
__global__ void k_pack_w(const float* __restrict__ Wmu, const float* __restrict__ Wlv,
                         float* __restrict__ Wcat) {
    int idx = blockIdx.x * blockDim.x + threadIdx.x;   // 0..16383
    int k = idx >> 7, c = idx & 127;
    Wcat[idx] = (c < 64) ? Wmu[k * 64 + c] : Wlv[k * 64 + (c - 64)];
}

// ---------------------------------------------------------------------------
// C[nrows x 128] = A[nrows x 128] @ W[128 x 128] via V_WMMA_F32_16X16X4_F32.
// Block = 256 threads = 8 waves. Block b owns rows [16b,16b+16); wave w owns
// cols [16w,16w+16). K-loop steps by 4 (fp32 WMMA K=4).
//
// A-frag (16x4 f32, 2 VGPR): lane L -> row L%16, K-pair (L<16 ? {0,1}:{2,3})
// B-frag (4x16 f32, 2 VGPR): lane L -> col L%16, same K-pair split
// C/D (16x16 f32, 8 VGPR):   lane L -> col L%16, row = vgpr + (L<16?0:8)
// ---------------------------------------------------------------------------
__global__ void k_gemm128(const float* __restrict__ A, const float* __restrict__ W,
                          float* __restrict__ C, int nrows) {
    const int lane  = threadIdx.x & 31;
    const int wave  = threadIdx.x >> 5;       // 0..7 -> column tile
    const int row0  = blockIdx.x * 16;
    const int col0  = wave * 16;
    const int m     = lane & 15;              // row index (A) / col index (B,C)
    const int khalf = (lane < 16) ? 0 : 2;

    const float* __restrict__ arow = A + (size_t)(row0 + m) * 128;
    v8f acc = {};
#pragma unroll
    for (int k = 0; k < 128; k += 4) {
        v2f a, b;
        const float2 av = *reinterpret_cast<const float2*>(arow + k + khalf);
        a.x = av.x;
        a.y = av.y;
        b.x = W[(size_t)(k + khalf) * 128 + col0 + m];
        b.y = W[(size_t)(k + khalf + 1) * 128 + col0 + m];
        acc = __builtin_amdgcn_wmma_f32_16x16x4_f32(
            /*neg_a=*/false, a, /*neg_b=*/false, b,
            /*c_mod=*/(short)0, acc, /*reuse_a=*/false, /*reuse_b=*/false);
    }
    const int rbase = (lane < 16) ? 0 : 8;
#pragma unroll
    for (int j = 0; j < 8; ++j) {
        C[(size_t)(row0 + rbase + j) * 128 + col0 + m] = acc[j];
    }
}

// ---------------------------------------------------------------------------
// Edge scatter: one wave per edge; 32 lanes x float4 cover 128 channels.
// out[dst] += norm * xw[src].   SPLIT=true routes cols [0,64) to the mu block
// of d_out and cols [64,128) to the logvar block (each N x 64, row-major).
// ---------------------------------------------------------------------------
template <bool SPLIT>
__global__ void k_scatter(const float* __restrict__ xw, const int* __restrict__ src,
                          const int* __restrict__ dst, const float* __restrict__ dis,
                          float* __restrict__ out, int E, int N) {
    const int gid  = blockIdx.x * blockDim.x + threadIdx.x;
    const int e    = gid >> 5;
    const int lane = gid & 31;
    if (e >= E) return;
    const int s = src[e];
    const int d = dst[e];
    const float nrm = dis[s] * dis[d];
    const int c = lane * 4;
    const float4 v = *reinterpret_cast<const float4*>(xw + (size_t)s * 128 + c);
    float* p;
    if (!SPLIT) {
        p = out + (size_t)d * 128 + c;
    } else {
        p = (c < 64) ? out + (size_t)d * 64 + c
                     : out + (size_t)N * 64 + (size_t)d * 64 + (c - 64);
    }
    unsafeAtomicAdd(p + 0, v.x * nrm);
    unsafeAtomicAdd(p + 1, v.y * nrm);
    unsafeAtomicAdd(p + 2, v.z * nrm);
    unsafeAtomicAdd(p + 3, v.w * nrm);
}

// h = relu(acc + dis^2 * xw + b1), in place over acc (N x 128)
__global__ void k_selfloop_relu(float* __restrict__ acc, const float* __restrict__ xw,
                                const float* __restrict__ dis, const float* __restrict__ b,
                                int N) {
    size_t idx = (size_t)blockIdx.x * blockDim.x + threadIdx.x;
    if (idx >= (size_t)N * 128) return;
    int i = (int)(idx >> 7);
    int c = (int)(idx & 127);
    float di = dis[i];
    float v  = acc[idx] + di * di * xw[idx] + b[c];
    acc[idx] = fmaxf(v, 0.0f);
}

// d_out(mu/lv blocks) += dis^2 * xwcat + bias   (self-loop + bias epilogue)
__global__ void k_final(float* __restrict__ out, const float* __restrict__ xwcat,
                        const float* __restrict__ dis, const float* __restrict__ bmu,
                        const float* __restrict__ blv, int N) {
    size_t idx = (size_t)blockIdx.x * blockDim.x + threadIdx.x;
    if (idx >= (size_t)N * 128) return;
    int i = (int)(idx >> 7);
    int c = (int)(idx & 127);
    float di  = dis[i];
    float val = di * di * xwcat[idx];
    if (c < 64) {
        out[(size_t)i * 64 + c] += val + bmu[c];
    } else {
        out[(size_t)N * 64 + (size_t)i * 64 + (c - 64)] += val + blv[c - 64];
    }
}

// ---------------------------------------------------------------------------
extern "C" void kernel_launch(void* const* d_in, const int* in_sizes, int n_in,
                              void* d_out, int out_size, void* d_ws, size_t ws_size,
                              hipStream_t stream) {
    const float* x    = (const float*)d_in[0];
    const int*   edge = (const int*)d_in[1];   // [2, E]
    // d_in[2] = batch (unused by the reference math)
    const float* W1   = (const float*)d_in[3];
    const float* b1   = (const float*)d_in[4];
    const float* Wmu  = (const float*)d_in[5];
    const float* bmu  = (const float*)d_in[6];
    const float* Wlv  = (const float*)d_in[7];
    const float* blv  = (const float*)d_in[8];

    const int N = in_sizes[0] / 128;           // 100000 (multiple of 16)
    const int E = in_sizes[1] / 2;             // 800000
    const int* src = edge;
    const int* dst = edge + E;

    // Workspace carve-up (floats): deg | dis | Wcat | B1 | B2
    float* ws   = (float*)d_ws;
    float* deg  = ws;
    float* dis  = deg + N;
    float* Wcat = dis + N;
    float* B1   = Wcat + 128 * 128;            // xw1, later xwcat   (N x 128)
    float* B2   = B1 + (size_t)N * 128;        // acc1, later h      (N x 128)
    float* outp = (float*)d_out;

    const size_t NC = (size_t)N * 128;
    const int nThreads = 256;
    const int gFill    = 2048;
    const int gE       = (E + nThreads - 1) / nThreads;
    const int gN       = (N + nThreads - 1) / nThreads;
    const int gNC      = (int)((NC + nThreads - 1) / nThreads);
    const int gEdgeW   = (int)(((size_t)E * 32 + nThreads - 1) / nThreads);

    // Degrees + symmetric normalization
    k_fill<<<gFill, nThreads, 0, stream>>>(deg, (size_t)N, 1.0f);   // self-loop
    k_deg_edges<<<gE, nThreads, 0, stream>>>(dst, deg, E);
    k_dis<<<gN, nThreads, 0, stream>>>(deg, dis, N);
    k_pack_w<<<64, nThreads, 0, stream>>>(Wmu, Wlv, Wcat);

    // Layer 1: xw1 = x @ W1 ; scatter ; h = relu(acc + dis^2*xw1 + b1)
    k_gemm128<<<N / 16, nThreads, 0, stream>>>(x, W1, B1, N);
    k_fill<<<gFill, nThreads, 0, stream>>>(B2, NC, 0.0f);
    k_scatter<false><<<gEdgeW, nThreads, 0, stream>>>(B1, src, dst, dis, B2, E, N);
    k_selfloop_relu<<<gNC, nThreads, 0, stream>>>(B2, B1, dis, b1, N);

    // Layer 2 (mu & logvar fused): xwcat = h @ [Wmu|Wlv] ; scatter into d_out
    k_gemm128<<<N / 16, nThreads, 0, stream>>>(B2, Wcat, B1, N);
    k_fill<<<gFill, nThreads, 0, stream>>>(outp, NC, 0.0f);
    k_scatter<true><<<gEdgeW, nThreads, 0, stream>>>(B1, src, dst, dis, outp, E, N);
    k_final<<<gNC, nThreads, 0, stream>>>(outp, B1, dis, bmu, blv, N);
}